// ChromosomeEmbedding_37503654429066
// MI455X (gfx1250) — compile-verified
//
#include <hip/hip_runtime.h>
#include <stdint.h>

// ChromosomeEmbedding broadcast: out[b, bin, :] = ce[chr[b]-1, :]
// BS=512, BINS=2001, DIM=128  -> 512 MB of pure stores, 0 FLOPs.
// Primary path: CDNA5 Tensor Data Mover iterate-store (replicate LDS tile
// across global memory). Fallback: non-temporal B128 streaming stores.

#define BS    512
#define BINS  2001
#define DIM   128
#define ROW_BYTES (DIM * 4)

typedef __attribute__((ext_vector_type(4))) unsigned int u32x4;
typedef __attribute__((ext_vector_type(8))) int          i32x8;
typedef __attribute__((ext_vector_type(4))) int          i32x4;
typedef __attribute__((ext_vector_type(4))) float        f32x4;  // native clang vector
                                                                 // (HIP float4 is a struct and
                                                                 //  rejected by nontemporal builtins)

#if defined(__HIP_DEVICE_COMPILE__) && __has_builtin(__builtin_amdgcn_tensor_store_from_lds)
#define USE_TDM 1
#else
#define USE_TDM 0
#endif

#if USE_TDM

__device__ __forceinline__ void tdm_store(u32x4 g0, i32x8 g1, i32x4 g2, i32x4 g3) {
#if defined(__clang_major__) && (__clang_major__ >= 23)
    // amdgpu-toolchain (clang-23 / therock-10.0): 6-arg form
    i32x8 g4 = {};
    __builtin_amdgcn_tensor_store_from_lds(g0, g1, g2, g3, g4, 0);
#else
    // ROCm 7.2 (clang-22): 5-arg form
    __builtin_amdgcn_tensor_store_from_lds(g0, g1, g2, g3, 0);
#endif
}

// One wave per (sample, bin-chunk). Wave stages the gathered 512B row into
// LDS, then fires TENSOR_STORE_FROM_LDS ops with iterate_enable=1 so the TDM
// engine replicates the row across up to 256 bins per instruction.
__global__ __launch_bounds__(32) void chrembed_tdm(const int* __restrict__ chr,
                                                   const float* __restrict__ ce,
                                                   float* __restrict__ out) {
    __shared__ __align__(16) float row[DIM];

    const int b    = blockIdx.x;           // sample
    const int y    = blockIdx.y;           // bin-chunk (0..3), 512 bins each
    const int lane = threadIdx.x;          // 0..31

    const int r = chr[b] - 1;              // scalar (uniform) gather index
    ((f32x4*)row)[lane] = ((const f32x4*)(ce + (size_t)r * DIM))[lane];
    asm volatile("s_wait_dscnt 0x0" ::: "memory");  // LDS visible before TDM reads it

    const unsigned lds_addr = (unsigned)(uintptr_t)row;   // low 32 bits = LDS byte addr
    const uint64_t base = (uint64_t)(uintptr_t)(out + (size_t)b * BINS * DIM);

    const int bin_lo = y * 512;
    int bin_hi = bin_lo + 512; if (bin_hi > BINS) bin_hi = BINS;

    for (int start = bin_lo; start < bin_hi; start += 256) {
        int iters = bin_hi - start; if (iters > 256) iters = 256;
        const uint64_t ga = base + (uint64_t)start * ROW_BYTES;

        // ---- D# group 0 (128b): count=1 | lds_addr | global_addr | type=2
        u32x4 g0;
        g0[0] = 1u;                                           // count=1, user mode
        g0[1] = lds_addr;                                     // bits 63:32
        g0[2] = (unsigned)(ga & 0xFFFFFFFFull);               // global_addr[31:0]
        g0[3] = ((unsigned)(ga >> 32) & 0x01FFFFFFu)          // global_addr[56:32]
                | (2u << 30);                                 // type = "image"

        // ---- D# group 1 (256b)
        i32x8 g1;
        g1[0] = (int)((2u << 16) | (1u << 19)); // data_size=4B, iterate_enable=1, mask=0
        g1[1] = (int)((unsigned)DIM  << 16);    // tensor_dim0[15:0]=128 (bits 63:48)
        g1[2] = (int)((unsigned)BINS << 16);    // tensor_dim1[15:0]=2001 (bits 95:80)
        g1[3] = (int)((unsigned)DIM  << 16);    // tile_dim0=128 (bits 127:112)
        g1[4] = 1;                              // tile_dim1=1, tile_dim2=0
        g1[5] = DIM;                            // tensor_dim0_stride[31:0]=128
        g1[6] = (int)((unsigned)DIM << 16);     // tensor_dim1_stride[15:0]=128
        g1[7] = 0;                              // tensor_dim1_stride[47:16]=0

        // ---- D# group 2 (128b), iterate mode
        i32x4 g2;
        g2[0] = 0;                              // tensor_dim2 (unused, 2D)
        g2[1] = 0;                              // lds_addr_increment = 0 (re-read row)
        g2[2] = DIM;                            // global_addr_increment = 128 elements
        g2[3] = (int)((unsigned)(iters - 1) << 16); // iterate_count (0 => 1x)

        // ---- D# group 3 (128b): unused dims
        i32x4 g3 = {};

        tdm_store(g0, g1, g2, g3);
    }

#if __has_builtin(__builtin_amdgcn_s_wait_tensorcnt)
    __builtin_amdgcn_s_wait_tensorcnt(0);
#else
    asm volatile("s_wait_tensorcnt 0x0" ::: "memory");
#endif
}

#endif // USE_TDM

// Fallback / reference data path: pure streaming stores.
// Lane layout: 32 lanes x f32x4 = one 128-float row per wave.
// Non-temporal: 512MB output >> 192MB L2 and is never re-read.
__global__ __launch_bounds__(256) void chrembed_store(const int* __restrict__ chr,
                                                      const float* __restrict__ ce,
                                                      float* __restrict__ out) {
    const int b     = blockIdx.x;
    const int lane  = threadIdx.x & 31;   // quad within row
    const int wv    = threadIdx.x >> 5;   // wave in block: 0..7
    const int nwv   = (blockDim.x >> 5) * gridDim.y;
    const int slot  = blockIdx.y * (blockDim.x >> 5) + wv;

    const int r = chr[b] - 1;
    const f32x4 v = ((const f32x4*)(ce + (size_t)r * DIM))[lane];

    f32x4* o = (f32x4*)out + (size_t)b * BINS * (DIM / 4) + lane;
    for (int bin = slot; bin < BINS; bin += nwv) {
        __builtin_nontemporal_store(v, o + (size_t)bin * (DIM / 4));
    }
}

extern "C" void kernel_launch(void* const* d_in, const int* in_sizes, int n_in,
                              void* d_out, int out_size, void* d_ws, size_t ws_size,
                              hipStream_t stream) {
    // setup_inputs order: tensor (f32, unused - only its shape matters),
    //                     chr (int32), ce (f32 [24,128])
    const int*   chr = (const int*)d_in[1];
    const float* ce  = (const float*)d_in[2];
    float*       out = (float*)d_out;
    (void)in_sizes; (void)n_in; (void)out_size; (void)d_ws; (void)ws_size;

#if __has_builtin(__builtin_amdgcn_tensor_store_from_lds)
    // 512 samples x 4 bin-chunks; one wave each issuing <=2 TDM iterate-stores.
    chrembed_tdm<<<dim3(BS, 4), 32, 0, stream>>>(chr, ce, out);
#else
    chrembed_store<<<dim3(BS, 4), 256, 0, stream>>>(chr, ce, out);
#endif
}